// FullyLoss_44805098832397
// MI455X (gfx1250) — compile-verified
//
#include <hip/hip_runtime.h>
#include <hip/hip_bf16.h>

typedef float v2f __attribute__((ext_vector_type(2)));
typedef float v4f __attribute__((ext_vector_type(4)));
typedef float v8f __attribute__((ext_vector_type(8)));

#define LOG_CLAMP (-100.0f)

// Wave32 sum-reduction using the matrix pipe:
// A (16x4 f32) holds the 32 per-lane partials in its K=0 / K=2 slots
// (A VGPR0: lanes 0-15 -> (M=lane,K=0), lanes 16-31 -> (M=lane-16,K=2)),
// A VGPR1 = 0.  B = all-ones (layout-agnostic since every element is 1).
// D[m][n] = acc[m] + acc[m+16] for all n.  Summing the 8 C VGPRs per lane
// gives S_low on lanes 0-15 and S_high on lanes 16-31; one xor-shuffle by 16
// yields the full wave sum in every lane.
__device__ __forceinline__ float wave_reduce32(float x) {
    v2f a; a[0] = x;    a[1] = 0.0f;
    v2f b; b[0] = 1.0f; b[1] = 1.0f;
    v8f c = {};
    c = __builtin_amdgcn_wmma_f32_16x16x4_f32(
            /*neg_a=*/false, a, /*neg_b=*/false, b,
            /*c_mod=*/(short)0, c, /*reuse_a=*/false, /*reuse_b=*/false);
    float s = ((c[0] + c[1]) + (c[2] + c[3])) + ((c[4] + c[5]) + (c[6] + c[7]));
    s += __shfl_xor(s, 16, 32);
    return s;
}

// One block handles ROWS_PER_BLOCK consecutive rows.  256 threads stream the
// row with 128-bit non-temporal loads (data is touched exactly once; keep it
// out of the way of L2 reuse).  BCE per element with torch-style clamping:
//   bce = -(y*log(p) + (1-y)*log(1-p)),  logs clamped at -100.
__global__ __launch_bounds__(256) void bce_partial_kernel(
    const float* __restrict__ outp, const float* __restrict__ labp,
    const int* __restrict__ srcp, float* __restrict__ partial,
    int B, int C4, int rows_per_block)
{
    const int tid  = threadIdx.x;
    const int row0 = blockIdx.x * rows_per_block;

    float acc = 0.0f;
    for (int r = 0; r < rows_per_block; ++r) {
        const int row = row0 + r;
        if (row >= B) break;                       // uniform across the block
        const int   s = srcp[row];                 // scalar (block-uniform) load
        const float m = (s == 0 || s == 1) ? 1.0f : 0.0f;

        const v4f* po = (const v4f*)outp + (size_t)row * C4;
        const v4f* pl = (const v4f*)labp + (size_t)row * C4;

        float e = 0.0f;
        for (int c = tid; c < C4; c += 256) {
            const v4f o = __builtin_nontemporal_load(po + c);
            const v4f y = __builtin_nontemporal_load(pl + c);
#pragma unroll
            for (int k = 0; k < 4; ++k) {
                // log1p(-p) == log(1-p): exact subtraction for p>=0.5
                // (Sterbenz), negligible rounding below; log(0) -> -inf is
                // absorbed by the clamp.
                const float lp = fmaxf(__logf(o[k]),        LOG_CLAMP);
                const float lq = fmaxf(__logf(1.0f - o[k]), LOG_CLAMP);
                e -= y[k] * (lp - lq) + lq;        // = -(y*lp + (1-y)*lq)
            }
        }
        acc += m * e;
    }

    // intra-wave reduction on the matrix pipe (EXEC is all-1s here: every
    // thread of the block reaches this point unconditionally)
    const float wsum = wave_reduce32(acc);

    __shared__ float sm[8];
    const int lane = tid & 31;
    const int wave = tid >> 5;
    if (lane == 0) sm[wave] = wsum;
    __syncthreads();
    if (tid == 0) {
        float t = 0.0f;
        for (int i = 0; i < 8; ++i) t += sm[i];   // fixed order: deterministic
        partial[blockIdx.x] = t;
    }
}

// Single block, fixed-order deterministic final reduction + scale by 1/B.
__global__ __launch_bounds__(256) void bce_final_kernel(
    const float* __restrict__ partial, int n, float scale,
    float* __restrict__ out)
{
    const int tid = threadIdx.x;
    float acc = 0.0f;
    for (int i = tid; i < n; i += 256) acc += partial[i];

    const float wsum = wave_reduce32(acc);

    __shared__ float sm[8];
    const int lane = tid & 31;
    const int wave = tid >> 5;
    if (lane == 0) sm[wave] = wsum;
    __syncthreads();
    if (tid == 0) {
        float t = 0.0f;
        for (int i = 0; i < 8; ++i) t += sm[i];
        out[0] = t * scale;
    }
}

extern "C" void kernel_launch(void* const* d_in, const int* in_sizes, int n_in,
                              void* d_out, int out_size, void* d_ws, size_t ws_size,
                              hipStream_t stream) {
    const float* outp = (const float*)d_in[0];   // output [B, C] f32
    const float* labp = (const float*)d_in[1];   // label  [B, C] f32
    const int*   srcp = (const int*)d_in[2];     // source [B]    i32

    const int BC = in_sizes[0];
    const int B  = in_sizes[2];
    const int C  = BC / B;                       // 1024
    const int C4 = C / 4;                        // 256 float4 per row

    const int rows_per_block = 8;
    const int nblocks = (B + rows_per_block - 1) / rows_per_block;  // 2048

    float* partial = (float*)d_ws;               // nblocks * 4 bytes = 8 KB

    bce_partial_kernel<<<nblocks, 256, 0, stream>>>(
        outp, labp, srcp, partial, B, C4, rows_per_block);
    bce_final_kernel<<<1, 256, 0, stream>>>(
        partial, nblocks, 1.0f / (float)B, (float*)d_out);
}